// SGC4Content_55834574848374
// MI455X (gfx1250) — compile-verified
//
#include <hip/hip_runtime.h>

typedef __attribute__((ext_vector_type(2))) float v2f;
typedef __attribute__((ext_vector_type(8))) float v8f;

// ---------------------------------------------------------------------------
// GCN normalization helpers
// ---------------------------------------------------------------------------
__global__ void k_init_deg(float* __restrict__ deg, int n) {
  int i = blockIdx.x * blockDim.x + threadIdx.x;
  if (i < n) deg[i] = 1.0f;  // self-loop contributes 1
}

__global__ void k_accum_deg(float* __restrict__ deg,
                            const long long* __restrict__ dst, int e) {
  int i = blockIdx.x * blockDim.x + threadIdx.x;
  if (i < e) atomicAdd(&deg[(int)dst[i]], 1.0f);
}

__global__ void k_dinv(float* __restrict__ deg, int n) {
  int i = blockIdx.x * blockDim.x + threadIdx.x;
  if (i < n) {
    float d = deg[i];
    deg[i] = (d > 0.0f) ? rsqrtf(d) : 0.0f;  // in place: deg -> dinv
  }
}

// ---------------------------------------------------------------------------
// Feature padding 56 -> 64 (zeros in cols 56..63; preserved by propagation)
// ---------------------------------------------------------------------------
__global__ void k_pad_x(const float* __restrict__ x, float* __restrict__ h, int n) {
  int i = blockIdx.x * blockDim.x + threadIdx.x;  // n*64 threads
  if (i < n * 64) {
    int node = i >> 6, f = i & 63;
    h[i] = (f < 56) ? x[node * 56 + f] : 0.0f;
  }
}

__global__ void k_pad_w1(const float* __restrict__ w1, float* __restrict__ w1p) {
  int i = blockIdx.x * blockDim.x + threadIdx.x;  // 64*64 threads
  if (i < 64 * 64) {
    int r = i >> 6;
    w1p[i] = (r < 56) ? w1[i] : 0.0f;  // w1 row-major 56x64; same flat index for r<56
  }
}

// ---------------------------------------------------------------------------
// One propagation hop:  hout = dinv^2 * hin (self loop)  then edge scatter-add
// ---------------------------------------------------------------------------
__global__ void k_selfloop(const float* __restrict__ hin, float* __restrict__ hout,
                           const float* __restrict__ dinv, int n) {
  int i = blockIdx.x * blockDim.x + threadIdx.x;  // n*64 threads
  if (i < n * 64) {
    float d = dinv[i >> 6];
    hout[i] = d * d * hin[i];
  }
}

__global__ void k_edge_scatter(const float* __restrict__ hin, float* __restrict__ hout,
                               const float* __restrict__ dinv,
                               const long long* __restrict__ src,
                               const long long* __restrict__ dst, int e) {
  int i = blockIdx.x * blockDim.x + threadIdx.x;  // e*16 threads (4 feats each)
  if (i >= e * 16) return;
  int ed = i >> 4;
  int f4 = (i & 15) << 2;
  int s = (int)src[ed];
  int d = (int)dst[ed];
  float nrm = dinv[s] * dinv[d];
  float4 v = *reinterpret_cast<const float4*>(hin + (size_t)s * 64 + f4);
  float* o = hout + (size_t)d * 64 + f4;
  atomicAdd(o + 0, nrm * v.x);
  atomicAdd(o + 1, nrm * v.y);
  atomicAdd(o + 2, nrm * v.z);
  atomicAdd(o + 3, nrm * v.w);
}

// ---------------------------------------------------------------------------
// FP32 WMMA GEMM:  C[M,N] = A[M,64] * B[64,N] + bias  (one wave per 16x16 tile)
// N is a compile-time constant so every B load / C store is base + imm offset.
// A frag per ISA 16x4 layout: lane -> M = lane%16, K = 2*(lane/16)+{0,1}
// B/C/D row-striped: lane -> N = lane%16; C VGPR v -> M = v + 8*(lane/16)
// ---------------------------------------------------------------------------
template <int N>
__global__ void k_gemm_wmma(const float* __restrict__ A, const float* __restrict__ B,
                            const float* __restrict__ bias, float* __restrict__ C,
                            int M) {
  const int lane = threadIdx.x & 31;
  const int wave = threadIdx.x >> 5;
  const int tile = blockIdx.x * (blockDim.x >> 5) + wave;
  constexpr int tilesN = N >> 4;
  const int total = (M >> 4) * tilesN;
  if (tile >= total) return;  // wave-uniform exit; EXEC all-ones around WMMA
  const int tm = tile / tilesN;
  const int tn = tile - tm * tilesN;
  const int lm = lane & 15;
  const int hi = lane >> 4;
  const int kb = hi << 1;
  const int col = tn * 16 + lm;

  const float bv = bias[col];
  v8f c = {bv, bv, bv, bv, bv, bv, bv, bv};

  const float* arow = A + (size_t)(tm * 16 + lm) * 64;  // + imm k offsets
  const float* bcol = B + kb * N + col;                 // + imm k*N offsets
#pragma unroll
  for (int k = 0; k < 64; k += 4) {
    v2f a = *(const v2f*)(arow + k + kb);  // 8B-aligned global_load_b64
    v2f b;
    b[0] = bcol[k * N];
    b[1] = bcol[k * N + N];
    c = __builtin_amdgcn_wmma_f32_16x16x4_f32(false, a, false, b,
                                              (short)0, c, false, false);
  }

  float* crow = C + (size_t)(tm * 16 + hi * 8) * N + col;  // + imm v*N offsets
#pragma unroll
  for (int v = 0; v < 8; v++) crow[v * N] = c[v];
}

// ---------------------------------------------------------------------------
extern "C" void kernel_launch(void* const* d_in, const int* in_sizes, int n_in,
                              void* d_out, int out_size, void* d_ws, size_t ws_size,
                              hipStream_t stream) {
  const float* x      = (const float*)d_in[0];
  const long long* ei = (const long long*)d_in[1];
  const float* W1 = (const float*)d_in[2];
  const float* b1 = (const float*)d_in[3];
  const float* W2 = (const float*)d_in[4];
  const float* b2 = (const float*)d_in[5];
  const float* W3 = (const float*)d_in[6];
  const float* b3 = (const float*)d_in[7];
  const float* W4 = (const float*)d_in[8];
  const float* b4 = (const float*)d_in[9];
  float* out = (float*)d_out;

  const int N = in_sizes[0] / 56;   // 50000
  const int E = in_sizes[1] / 2;    // 800000
  const long long* srcp = ei;
  const long long* dstp = ei + E;

  // Workspace layout (floats): dinv | hA | hB | W1p
  float* ws = (float*)d_ws;
  float* dinv = ws;
  size_t off = ((size_t)N + 63) & ~(size_t)63;
  float* hA = ws + off;
  float* hB = hA + (size_t)N * 64;
  float* W1p = hB + (size_t)N * 64;

  dim3 blk(256);
  auto cdiv = [](long long a, long long b) { return (unsigned)((a + b - 1) / b); };

  k_init_deg<<<cdiv(N, 256), blk, 0, stream>>>(dinv, N);
  k_accum_deg<<<cdiv(E, 256), blk, 0, stream>>>(dinv, dstp, E);
  k_dinv<<<cdiv(N, 256), blk, 0, stream>>>(dinv, N);
  k_pad_x<<<cdiv((long long)N * 64, 256), blk, 0, stream>>>(x, hA, N);
  k_pad_w1<<<cdiv(64 * 64, 256), blk, 0, stream>>>(W1, W1p);

  auto hop3 = [&]() {
    for (int h = 0; h < 3; h++) {
      k_selfloop<<<cdiv((long long)N * 64, 256), blk, 0, stream>>>(hA, hB, dinv, N);
      k_edge_scatter<<<cdiv((long long)E * 16, 256), blk, 0, stream>>>(hA, hB, dinv,
                                                                       srcp, dstp, E);
      float* t = hA; hA = hB; hB = t;
    }
  };

  const unsigned g64  = cdiv((long long)(N / 16) * 4, 8);   // N=64:  8 waves/block
  const unsigned g256 = cdiv((long long)(N / 16) * 16, 8);  // N=256

  hop3();
  k_gemm_wmma<64><<<g64, blk, 0, stream>>>(hA, W1p, b1, hB, N);
  { float* t = hA; hA = hB; hB = t; }
  hop3();
  k_gemm_wmma<64><<<g64, blk, 0, stream>>>(hA, W2, b2, hB, N);
  { float* t = hA; hA = hB; hB = t; }
  hop3();
  k_gemm_wmma<64><<<g64, blk, 0, stream>>>(hA, W3, b3, hB, N);
  { float* t = hA; hA = hB; hB = t; }
  k_gemm_wmma<256><<<g256, blk, 0, stream>>>(hA, W4, b4, out, N);
}